// PCENAdaptation_Res_21844203667491
// MI455X (gfx1250) — compile-verified
//
#include <hip/hip_runtime.h>
#include <stdint.h>

#define EPSV 1e-6f
#define T_LEN 32768
#define ROWS 1024            // B*C = 8*128
#define PARTS 32
#define PART_T (T_LEN / PARTS)   // 1024
#define TILE_T 128
#define ITERS (PART_T / TILE_T)  // 8
#define LDS_STRIDE 132           // 128 + pad: 132 mod 64 = 4 -> conflict-free column reads

typedef __attribute__((ext_vector_type(2))) float v2f;
typedef __attribute__((ext_vector_type(4))) float v4f;
typedef __attribute__((ext_vector_type(8))) float v8f;

__device__ __forceinline__ float fexp2(float x) { return __builtin_exp2f(x); }
__device__ __forceinline__ float flog2(float x) { return __builtin_log2f(x); }

// Async copy 32B (2 x b128) global -> LDS; tracked by ASYNCcnt.
// Per CDNA5 ISA: INST_OFFSET applies to both LDS and global addr.
__device__ __forceinline__ void async_copy32(unsigned lds_addr, const float* gaddr) {
    asm volatile("global_load_async_to_lds_b128 %0, %1, off\n\t"
                 "global_load_async_to_lds_b128 %0, %1, off offset:16"
                 :: "v"(lds_addr), "v"(gaddr)
                 : "memory");
}
__device__ __forceinline__ void wait_async0() {
    asm volatile("s_wait_asynccnt 0x0" ::: "memory");
}

// ---------------------------------------------------------------------------
// Kernel 1: per-partition partial sums of u_t = relu(x_t) * d^t, per row.
// Grid: (PARTS, ROWS/16). Block: 256. 16 threads per row, float4 streaming.
// ---------------------------------------------------------------------------
__global__ __launch_bounds__(256) void pcen_partials(
    const float* __restrict__ x, const float* __restrict__ s_p,
    float* __restrict__ partials)
{
    const int p    = blockIdx.x;
    const int rblk = blockIdx.y;
    const int tid  = threadIdx.x;
    const int rr   = tid >> 4;     // row in tile, 0..15
    const int l16  = tid & 15;

    const float s   = fmaxf(s_p[0], 1e-5f);
    const float dd  = 1.0f - s;
    const float l2d = flog2(dd);

    const int row = rblk * 16 + rr;
    const float* __restrict__ xrow = x + (size_t)row * T_LEN + (size_t)p * PART_T;

    // decay recurrence: one exp2 up front, then *= d^64 per step
    float d0        = fexp2((float)(p * PART_T + l16 * 4) * l2d);
    const float d64 = fexp2(64.0f * l2d);

    float acc = 0.0f;
    for (int k = 0; k < PART_T; k += 64) {
        const int t = k + l16 * 4;
        v4f xv = *(const v4f*)(xrow + t);
        const float d1 = d0 * dd;
        const float d2 = d1 * dd;
        const float d3 = d2 * dd;
        acc += fmaxf(xv.x, 0.0f) * d0 + fmaxf(xv.y, 0.0f) * d1
             + fmaxf(xv.z, 0.0f) * d2 + fmaxf(xv.w, 0.0f) * d3;
        d0 *= d64;
    }
    // reduce across the 16 lanes of this row (stays within 16-lane halves)
    #pragma unroll
    for (int m = 1; m < 16; m <<= 1)
        acc += __shfl_xor(acc, m, 32);
    if (l16 == 0)
        partials[(size_t)row * PARTS + p] = acc;
}

// ---------------------------------------------------------------------------
// Kernel 2: exclusive scan of the 32 partition partials of each row.
// ---------------------------------------------------------------------------
__global__ __launch_bounds__(32) void pcen_scan_partials(
    const float* __restrict__ partials, float* __restrict__ offsets)
{
    const int row  = blockIdx.x;
    const int lane = threadIdx.x;
    const float v = partials[(size_t)row * PARTS + lane];
    float inc = v;
    #pragma unroll
    for (int d = 1; d < 32; d <<= 1) {
        float t = __shfl_up(inc, d, 32);
        if (lane >= d) inc += t;
    }
    offsets[(size_t)row * PARTS + lane] = inc - v;   // exclusive prefix
}

// ---------------------------------------------------------------------------
// Kernel 3: local inclusive scan via V_WMMA_F32_16X16X4_F32 + pointwise PCEN.
// Double-buffered async global->LDS staging overlapped with compute.
// Grid: (PARTS, ROWS/16). Block: 256 (8 waves).
// ---------------------------------------------------------------------------
__global__ __launch_bounds__(256) void pcen_apply(
    const float* __restrict__ x,
    const float* __restrict__ alpha_p, const float* __restrict__ delta_p,
    const float* __restrict__ r_p,     const float* __restrict__ s_p,
    const float* __restrict__ offsets,
    float* __restrict__ y)
{
    __shared__ __align__(16) float xs[2][16 * LDS_STRIDE];
    __shared__ __align__(16) float ys[16 * LDS_STRIDE];
    __shared__ float totals[8][16];
    __shared__ float carry[16];

    const int p    = blockIdx.x;
    const int rblk = blockIdx.y;
    const int tid  = threadIdx.x;
    const int w    = tid >> 5;      // wave id 0..7  -> time chunk
    const int lane = tid & 31;
    const int n    = lane & 15;     // WMMA column (row of the signal tile)
    const int hi   = lane >> 4;     // lane half: selects K pair / M half
    const int rr   = tid >> 4;      // staging row 0..15
    const int l16  = tid & 15;

    const float alpha = alpha_p[0];
    const float delta = delta_p[0];
    const float r     = r_p[0];
    const float s     = fmaxf(s_p[0], 1e-5f);
    const float dd    = 1.0f - s;
    const float l2d   = flog2(dd);
    const float inv_s = 1.0f / s;
    const float dpr   = fexp2(r * flog2(delta));   // delta^r

    // constant powers of d (constant-indexed after unroll -> registers)
    const float dsq = dd * dd;
    const float d4  = dsq * dsq;
    const float d8  = d4 * d4;
    const float c4[4] = {1.0f, d4, d8, d8 * d4};   // d^(4i) for B frags
    float cv[10];                                   // d^j, j=0..9 for pointwise
    cv[0] = 1.0f;
    #pragma unroll
    for (int j = 1; j <= 9; ++j) cv[j] = cv[j - 1] * dd;
    const float d2h = hi ? dsq : 1.0f;             // d^(2*hi)
    const float d8h = hi ? d8  : 1.0f;             // d^(8*hi)

    const size_t xbase = (size_t)(rblk * 16 + rr) * T_LEN + (size_t)p * PART_T;
    const unsigned lds_x0 = (unsigned)(uintptr_t)&xs[0][rr * LDS_STRIDE + l16 * 8];
    const unsigned lds_x1 = (unsigned)(uintptr_t)&xs[1][rr * LDS_STRIDE + l16 * 8];

    // prologue: kick off tile 0, init carry
    async_copy32(lds_x0, x + xbase + l16 * 8);
    if (tid < 16)
        carry[tid] = offsets[(size_t)(rblk * 16 + tid) * PARTS + p];

    int cur = 0;
    for (int it = 0; it < ITERS; ++it) {
        wait_async0();        // this wave's async writes to xs[cur] are done
        __syncthreads();      // -> all waves' writes done: xs[cur] complete

        if (it + 1 < ITERS)   // overlap next tile's DMA with this tile's compute
            async_copy32(cur ? lds_x0 : lds_x1,
                         x + xbase + (it + 1) * TILE_T + l16 * 8);

        const float* xcur = xs[cur];
        const int tbase = p * PART_T + it * TILE_T + w * 16;   // global t of chunk start
        const float dchunk = fexp2((float)tbase * l2d);        // d^tbase (1 exp2/tile)

        // ---- WMMA prefix scan of this wave's 16-time chunk ----
        v8f acc = {};
        #pragma unroll
        for (int k4 = 0; k4 < 16; k4 += 4) {
            const int k0 = k4 + hi * 2;            // K pair held by this lane half
            v2f a, b;
            // A = lower-triangular ones: A[m][j] = (j <= m), m = lane&15
            a.x = (k0     <= n) ? 1.0f : 0.0f;
            a.y = (k0 + 1 <= n) ? 1.0f : 0.0f;
            // B[k][n] = relu(x) * d^(tbase+k), decay via hoisted powers
            const float dk0 = dchunk * c4[k4 >> 2] * d2h;
            const float x0  = fmaxf(xcur[n * LDS_STRIDE + (w * 16 + k0)],     0.0f);
            const float x1  = fmaxf(xcur[n * LDS_STRIDE + (w * 16 + k0 + 1)], 0.0f);
            b.x = x0 * dk0;
            b.y = x1 * (dk0 * dd);
            acc = __builtin_amdgcn_wmma_f32_16x16x4_f32(
                      false, a, false, b, (short)0, acc, false, false);
        }

        // chunk totals = D[m=15][n]  (VGPR 7 of the hi lane half)
        if (hi)
            totals[w][n] = acc[7];
        __syncthreads();

        // ---- cross-chunk carry: branch-free masked sum of preceding chunks ----
        float off = carry[n];
        #pragma unroll
        for (int ww = 0; ww < 7; ++ww) {
            const float tw = totals[ww][n];        // uniform unrolled LDS loads
            off += (ww < w) ? tw : 0.0f;           // v_cndmask, no divergence
        }

        // ---- pointwise PCEN on scanned values ----
        #pragma unroll
        for (int v = 0; v < 8; ++v) {
            const int m = v + hi * 8;              // time within chunk
            const float S    = acc[v] + off;
            const float wc   = (1.0f - dchunk * d8h * cv[v + 1]) * inv_s; // sum d^k, k<=tg
            const float M    = S / (wc + EPSV);
            const float xv   = fmaxf(xcur[n * LDS_STRIDE + w * 16 + m], 0.0f);
            const float norm = xv * fexp2(-alpha * flog2(EPSV + M));
            ys[n * LDS_STRIDE + w * 16 + m] =
                fexp2(r * flog2(norm + delta)) - dpr;
        }
        __syncthreads();      // ys ready; all reads of xs[cur] finished

        // ---- coalesced store + carry update ----
        {
            const float* srcl = &ys[rr * LDS_STRIDE + l16 * 8];
            v4f a = *(const v4f*)srcl;
            v4f b = *(const v4f*)(srcl + 4);
            float* dst = y + xbase + it * TILE_T + l16 * 8;
            *(v4f*)dst = a; *(v4f*)(dst + 4) = b;
        }
        if (tid < 16) {
            float c = carry[tid];
            #pragma unroll
            for (int ww = 0; ww < 8; ++ww)
                c += totals[ww][tid];
            carry[tid] = c;
        }
        cur ^= 1;
    }
}

// ---------------------------------------------------------------------------
extern "C" void kernel_launch(void* const* d_in, const int* in_sizes, int n_in,
                              void* d_out, int out_size, void* d_ws, size_t ws_size,
                              hipStream_t stream) {
    const float* x     = (const float*)d_in[0];
    const float* alpha = (const float*)d_in[1];
    const float* delta = (const float*)d_in[2];
    const float* r     = (const float*)d_in[3];
    const float* s     = (const float*)d_in[4];
    float* out = (float*)d_out;

    float* partials = (float*)d_ws;                    // ROWS*PARTS f32 = 128 KB
    float* offsets  = partials + (size_t)ROWS * PARTS; // another 128 KB

    dim3 grid(PARTS, ROWS / 16);   // 32 x 64
    pcen_partials<<<grid, 256, 0, stream>>>(x, s, partials);
    pcen_scan_partials<<<ROWS, 32, 0, stream>>>(partials, offsets);
    pcen_apply<<<grid, 256, 0, stream>>>(x, alpha, delta, r, s, offsets, out);
}